// LowRankSparseAttention_82240033784117
// MI455X (gfx1250) — compile-verified
//
#include <hip/hip_runtime.h>
#include <cstdint>

// ---------------------------------------------------------------------------
// LowRankSparseAttention for MI455X (gfx1250, wave32, WMMA)
// B=2, L=S=2048, D_MODEL=1024, H=16, RANK=32, TOPK=32, D_HEAD=64
// ---------------------------------------------------------------------------

typedef __attribute__((ext_vector_type(2))) float v2f;
typedef __attribute__((ext_vector_type(8))) float v8f;
typedef int v4i_ __attribute__((vector_size(16)));  // matches async-LDS builtin

#define BB_  2
#define LL_  2048
#define SS_  2048
#define DD_  1024
#define HH_  16
#define RR_  32
#define DH_  64
#define TOPK_ 32
#define MAXSEL 64

#if defined(__has_builtin)
#if __has_builtin(__builtin_amdgcn_global_load_async_to_lds_b128)
#define HAVE_ASYNC_LDS 1
#endif
#endif
#ifndef HAVE_ASYNC_LDS
#define HAVE_ASYNC_LDS 0
#endif

typedef v4i_ __attribute__((address_space(1))) *gvec_p;
typedef v4i_ __attribute__((address_space(3))) *lvec_p;

__device__ __forceinline__ v8f wmma_f32(v2f a, v2f b, v8f c) {
  // V_WMMA_F32_16X16X4_F32: (neg_a, A, neg_b, B, c_mod, C, reuse_a, reuse_b)
  return __builtin_amdgcn_wmma_f32_16x16x4_f32(false, a, false, b, (short)0, c,
                                               false, false);
}

// monotonic float <-> uint key (larger key <=> larger float)
__device__ __forceinline__ unsigned f2key(float f) {
  unsigned u = __float_as_uint(f);
  return (u & 0x80000000u) ? ~u : (u | 0x80000000u);
}
__device__ __forceinline__ float key2f(unsigned k) {
  unsigned u = (k & 0x80000000u) ? (k & 0x7fffffffu) : ~k;
  return __uint_as_float(u);
}

__device__ __forceinline__ unsigned wred_maxu(unsigned v) {
#pragma unroll
  for (int m = 16; m; m >>= 1) {
    unsigned o = (unsigned)__shfl_xor((int)v, m, 32);
    v = (o > v) ? o : v;
  }
  return v;
}
__device__ __forceinline__ int wred_sumi(int v) {
#pragma unroll
  for (int m = 16; m; m >>= 1) v += __shfl_xor(v, m, 32);
  return v;
}

// ---------------------------------------------------------------------------
// f32 WMMA GEMM: out = A[MxK] * Bw[KxN] + bias[N]
// block = 128 thr (4 waves) -> macro tile 16M x 128N.
// Wave w owns two 16x16 tiles: n = ny+w*16 and n = ny+w*16+64 -> each A
// fragment feeds 2 WMMAs.  A k-slab (16x32) is staged in LDS (async if the
// toolchain exposes GLOBAL_LOAD_ASYNC_TO_LDS), shared by all 4 waves.
// MODE 0: out row-major [M x N]
// MODE 1: scatter to Q   [B,H,L,R]
// MODE 2: scatter to K^T [B,H,R,S]
// MODE 3: scatter to V   [B,H,S,64]
// ---------------------------------------------------------------------------
#define AS_STRIDE 36  // 144B row stride: 16B-aligned rows, conflict-free reads

template <int MODE>
__global__ void gemm16_wmma(const float* __restrict__ A,
                            const float* __restrict__ Bw,
                            const float* __restrict__ bias,
                            float* __restrict__ out, int M, int N, int K) {
  __shared__ float As[16 * AS_STRIDE];

  const int lane = threadIdx.x & 31;
  const int wave = threadIdx.x >> 5;
  const int half = lane >> 4;   // 0: K pair {0,1}, 1: K pair {2,3}
  const int l16  = lane & 15;
  const int m0 = blockIdx.x * 16;
  const int nA = blockIdx.y * 128 + wave * 16;
  const int nB = nA + 64;

  // cooperative A-slab fill pattern: thread t -> row t/8, cols (t%8)*4..+3
  const int arow_c = threadIdx.x >> 3;
  const int acol_c = (threadIdx.x & 7) * 4;
  const float* gsrc = A + (size_t)(m0 + arow_c) * K + acol_c;
  float* ldst = &As[arow_c * AS_STRIDE + acol_c];

  v8f c0 = {0.f, 0.f, 0.f, 0.f, 0.f, 0.f, 0.f, 0.f};
  v8f c1 = c0;

  for (int k = 0; k < K; k += 32) {
    // ---- stage A[16][32] k-slab into LDS ----
#if HAVE_ASYNC_LDS
    __builtin_amdgcn_global_load_async_to_lds_b128(
        (gvec_p)(gsrc + k), (lvec_p)ldst, 0, 0);
    if (k + 32 < K) __builtin_prefetch(gsrc + k + 32, 0, 0);
    asm volatile("s_wait_asynccnt 0x0" ::: "memory");
    __syncthreads();
#else
    *(float4*)ldst = *(const float4*)(gsrc + k);
    if (k + 32 < K) __builtin_prefetch(gsrc + k + 32, 0, 0);
    __syncthreads();
#endif

    // ---- 8 WMMA k-steps ----
#pragma unroll
    for (int kk = 0; kk < 8; ++kk) {
      const float* ap = &As[l16 * AS_STRIDE + kk * 4 + half * 2];
      v2f a, b0, b1;
      a[0] = ap[0];                                 // lanes 0-15: K=k4,k4+1
      a[1] = ap[1];                                 // lanes 16-31: K=k4+2,k4+3
      const int kb = k + kk * 4 + half * 2;
      b0[0] = Bw[(size_t)kb * N + nA + l16];
      b0[1] = Bw[(size_t)(kb + 1) * N + nA + l16];
      b1[0] = Bw[(size_t)kb * N + nB + l16];
      b1[1] = Bw[(size_t)(kb + 1) * N + nB + l16];
      c0 = wmma_f32(a, b0, c0);
      c1 = wmma_f32(a, b1, c1);
    }
    __syncthreads();
  }

  const float biasA = bias[nA + l16];
  const float biasB = bias[nB + l16];
#pragma unroll
  for (int t = 0; t < 2; ++t) {
    const int n = (t ? nB : nA) + l16;
    const float bv = t ? biasB : biasA;
#pragma unroll
    for (int i = 0; i < 8; ++i) {
      const int m = m0 + i + half * 8;              // C/D: VGPR i = M row
      const float val = (t ? c1[i] : c0[i]) + bv;   //      lane(&15) = N col
      size_t idx;
      if (MODE == 0) {
        idx = (size_t)m * N + n;
      } else if (MODE == 1) {
        const int b_ = m / LL_, l = m % LL_, h = n >> 5, r = n & 31;
        idx = (((size_t)(b_ * HH_ + h) * LL_) + l) * RR_ + r;
      } else if (MODE == 2) {
        const int b_ = m / SS_, s = m % SS_, h = n >> 5, r = n & 31;
        idx = (((size_t)(b_ * HH_ + h) * RR_) + r) * SS_ + s;
      } else {
        const int b_ = m / SS_, s = m % SS_, h = n >> 6, d = n & 63;
        idx = (((size_t)(b_ * HH_ + h) * SS_) + s) * DH_ + d;
      }
      out[idx] = val;
    }
  }
}

// ---------------------------------------------------------------------------
// Attention: per block, 16 query rows of one (b,h).
// Phase 1 (4 waves split S): scores[16 x 2048] = Q_tile(16x32) @ K^T via WMMA
// Phase 2 (4 rows/wave): integer-key radix-select of the 32nd-largest score,
//   ballot-compaction of the selected set (>= kth, reference tie semantics),
//   softmax over selected, 32-entry weighted gather of V -> ctx[B,L,1024]
// ---------------------------------------------------------------------------
__global__ void attn_topk(const float* __restrict__ Qh,
                          const float* __restrict__ KhT,
                          const float* __restrict__ Vh,
                          const float* __restrict__ pos_bias,
                          float* __restrict__ ctx) {
  extern __shared__ float smem[];
  float* sc    = smem;                               // 16 * 2048 floats (128KB)
  int*   lists = (int*)(smem + 16 * SS_);            // 4 * MAXSEL ints
  float* lvals = (float*)(lists + 4 * MAXSEL);       // 4 * MAXSEL floats

  const int lane = threadIdx.x & 31;
  const int wave = threadIdx.x >> 5;
  const int half = lane >> 4;
  const int l16  = lane & 15;

  const int tiles_per_bh = LL_ / 16;                 // 128
  const int bh = blockIdx.x / tiles_per_bh;
  const int l0 = (blockIdx.x % tiles_per_bh) * 16;
  const int b  = bh / HH_;
  const int h  = bh % HH_;

  // ---------- phase 1: WMMA score strip ----------
  const float scale = 0.17677669529663688f;          // 1/sqrt(RANK)
  const float pb = pos_bias[h];

  {
    const float* qrow = Qh + ((size_t)bh * LL_ + l0 + l16) * RR_;
    float4 qf[8];
#pragma unroll
    for (int t = 0; t < 8; ++t) qf[t] = *(const float4*)(qrow + t * 4);

    for (int ct = wave * 32; ct < wave * 32 + 32; ++ct) {
      const int s0 = ct * 16;
      v8f c = {0.f, 0.f, 0.f, 0.f, 0.f, 0.f, 0.f, 0.f};
#pragma unroll
      for (int t = 0; t < 8; ++t) {
        v2f a, bf;
        a[0] = half ? qf[t].z : qf[t].x;
        a[1] = half ? qf[t].w : qf[t].y;
        const int r = t * 4 + half * 2;
        bf[0] = KhT[((size_t)bh * RR_ + r) * SS_ + s0 + l16];    // coalesced
        bf[1] = KhT[((size_t)bh * RR_ + r + 1) * SS_ + s0 + l16];
        c = wmma_f32(a, bf, c);
      }
#pragma unroll
      for (int i = 0; i < 8; ++i) {
        const int row = i + half * 8;
        sc[row * SS_ + s0 + l16] = c[i] * scale + pb;
      }
    }
  }
  __syncthreads();

  // ---------- phase 2: top-k + softmax + A@V ----------
  int*   wlist = lists + wave * MAXSEL;
  float* wval  = lvals + wave * MAXSEL;

  for (int rr = 0; rr < 4; ++rr) {
    const int row = wave * 4 + rr;
    const float* srow = sc + row * SS_;

    // convert once to ordered integer keys; all selection math is integer
    unsigned kr[64];
#pragma unroll
    for (int j = 0; j < 64; ++j) kr[j] = f2key(srow[j * 32 + lane]);

    unsigned mxk = 0u;
#pragma unroll
    for (int j = 0; j < 64; ++j) mxk = (kr[j] > mxk) ? kr[j] : mxk;
    const float mx = key2f(wred_maxu(mxk));

    // exact kth-largest via MSB-first radix search on keys
    unsigned prefix = 0u;
    for (int bit = 31; bit >= 0; --bit) {
      const unsigned cand = prefix | (1u << bit);
      int cnt = 0;
#pragma unroll
      for (int j = 0; j < 64; ++j) cnt += (kr[j] >= cand) ? 1 : 0;
      cnt = wred_sumi(cnt);
      if (cnt >= TOPK_) prefix = cand;
    }
    const unsigned kthkey = prefix;

    // ballot-compact indices with score >= kth (reference tie semantics)
    int cnt = 0;
#pragma unroll
    for (int j = 0; j < 64; ++j) {
      const bool p = (kr[j] >= kthkey);
      const unsigned mb = (unsigned)__ballot(p);
      const int pos = cnt + __popc(mb & ((1u << lane) - 1u));
      if (p && pos < MAXSEL) {
        wlist[pos] = j * 32 + lane;
        wval[pos]  = key2f(kr[j]);
      }
      cnt += __popc(mb);
    }
    if (cnt > MAXSEL) cnt = MAXSEL;
    // same-wave DS ops are in-order: list writes visible to reads below

    float acc0 = 0.f, acc1 = 0.f, wsum = 0.f;
    for (int i = 0; i < cnt; ++i) {
      const int s   = wlist[i];                      // LDS broadcast
      const float w = __expf(wval[i] - mx);
      wsum += w;                                     // identical in all lanes
      const float* vp = Vh + ((size_t)bh * SS_ + s) * DH_;
      acc0 += w * vp[lane];
      acc1 += w * vp[lane + 32];
    }
    const float inv = 1.0f / wsum;
    float* op = ctx + ((size_t)(b * LL_ + l0 + row)) * DD_ + h * DH_;
    op[lane]      = acc0 * inv;
    op[lane + 32] = acc1 * inv;
  }
}

// ---------------------------------------------------------------------------
extern "C" void kernel_launch(void* const* d_in, const int* in_sizes, int n_in,
                              void* d_out, int out_size, void* d_ws,
                              size_t ws_size, hipStream_t stream) {
  const float* q  = (const float*)d_in[0];
  const float* k  = (const float*)d_in[1];
  const float* v  = (const float*)d_in[2];
  const float* Wq = (const float*)d_in[3];
  const float* bq = (const float*)d_in[4];
  const float* Wk = (const float*)d_in[5];
  const float* bk = (const float*)d_in[6];
  const float* Wv = (const float*)d_in[7];
  const float* bv = (const float*)d_in[8];
  const float* Wo = (const float*)d_in[9];
  const float* bo = (const float*)d_in[10];
  const float* pb = (const float*)d_in[11];
  float* out = (float*)d_out;

  float* ws  = (float*)d_ws;
  float* Qh  = ws;                                         // [B,H,L,R]  8MB
  float* KhT = Qh  + (size_t)BB_ * HH_ * LL_ * RR_;        // [B,H,R,S]  8MB
  float* Vh  = KhT + (size_t)BB_ * HH_ * RR_ * SS_;        // [B,H,S,64] 16MB
  float* ctx = Vh  + (size_t)BB_ * HH_ * SS_ * DH_;        // [B,L,1024] 16MB

  const int M = BB_ * LL_;  // 4096
  const dim3 blk(128);

  // Q / K^T / V projections (f32 WMMA, LDS-staged A slabs)
  gemm16_wmma<1><<<dim3(M / 16, (HH_ * RR_) / 128), blk, 0, stream>>>(
      q, Wq, bq, Qh, M, HH_ * RR_, DD_);
  gemm16_wmma<2><<<dim3(M / 16, (HH_ * RR_) / 128), blk, 0, stream>>>(
      k, Wk, bk, KhT, M, HH_ * RR_, DD_);
  gemm16_wmma<3><<<dim3(M / 16, DD_ / 128), blk, 0, stream>>>(
      v, Wv, bv, Vh, M, DD_, DD_);

  // attention: 32 (b,h) * 128 query tiles = 4096 blocks, 130KB dynamic LDS
  const size_t lds_bytes =
      (size_t)16 * SS_ * sizeof(float) + 4 * MAXSEL * (sizeof(int) + sizeof(float));
  attn_topk<<<dim3(BB_ * HH_ * (LL_ / 16)), dim3(128), lds_bytes, stream>>>(
      Qh, KhT, Vh, pb, ctx);

  // output projection
  gemm16_wmma<0><<<dim3(M / 16, DD_ / 128), blk, 0, stream>>>(
      ctx, Wo, bo, out, M, DD_, DD_);
}